// QConv1d_3324304687602
// MI455X (gfx1250) — compile-verified
//
#include <hip/hip_runtime.h>
#include <math.h>

typedef __attribute__((ext_vector_type(2))) float v2f;
typedef __attribute__((ext_vector_type(8))) float v8f;

#define OUT_L 505
#define L_IN  512

// Composite of CNOT(a,a+1),(a+1,a+2),(a+2,a+3) on a 4-bit index (bit3 = MSB = first wire).
__device__ __forceinline__ int cnot_perm4(int r) {
  int b3 = (r >> 3) & 1, b2 = (r >> 2) & 1, b1 = (r >> 1) & 1, b0 = r & 1;
  int n1 = b2 ^ b3;        // new w(a+1)
  int n2 = b1 ^ n1;        // new w(a+2)
  int n3 = b0 ^ n2;        // new w(a+3)
  return (b3 << 3) | (n1 << 2) | (n2 << 1) | n3;
}

// Prepack K_A/K_B (Kronecker of 4 RY(theta/2) 2x2 rotations) per (channel, layer)
// directly in WMMA A-operand per-lane layout:
//   pack[((mat*4+q)*32 + lane)*2 + j] = K[lane%16][4q + 2*(lane/16) + j]
// mat = ch*8 + layer*2 + which   (which: 0 = wires0-3 (K_A), 1 = wires4-7 (K_B))
__global__ __launch_bounds__(256) void qconv_pack_kernel(const float* __restrict__ w,
                                                         float* __restrict__ pack) {
  int tid = blockIdx.x * blockDim.x + threadIdx.x;
  if (tid >= 64 * 32) return;
  int lane = tid & 31;
  int mat  = tid >> 5;
  int ch = mat >> 3, layer = (mat >> 1) & 3, which = mat & 1;
  const float* th = w + ch * 32 + layer * 8 + which * 4;
  float cc[4], ss[4];
#pragma unroll
  for (int t = 0; t < 4; ++t) {
    float h = th[t] * 0.5f;
    cc[t] = cosf(h);
    ss[t] = sinf(h);
  }
  int row = lane & 15, hi = lane >> 4;
#pragma unroll
  for (int q = 0; q < 4; ++q) {
#pragma unroll
    for (int j = 0; j < 2; ++j) {
      int col = 4 * q + 2 * hi + j;
      float v = 1.0f;
#pragma unroll
      for (int t = 0; t < 4; ++t) {
        int rb = (row >> (3 - t)) & 1;
        int cb = (col >> (3 - t)) & 1;
        // RY half-angle matrix [[c,-s],[s,c]]
        v *= rb ? (cb ? cc[t] : ss[t]) : (cb ? -ss[t] : cc[t]);
      }
      pack[(mat * 4 + q) * 64 + lane * 2 + j] = v;
    }
  }
}

// One wave32 per (batch, window, channel): state = 16x16 f32 matrix, layers via
// V_WMMA_F32_16X16X4_F32 chains; CNOT permutation folded into LDS scatter.
__global__ __launch_bounds__(256) void qconv_main_kernel(const float* __restrict__ x,
                                                         const float* __restrict__ pack,
                                                         float* __restrict__ out) {
  __shared__ float lds[8][2][256];          // per-wave S (col-major) and T (row-major)
  const int lane = threadIdx.x & 31;
  const int ch   = threadIdx.x >> 5;        // wave id in block = output channel
  const int ol   = blockIdx.x;
  const int b    = blockIdx.y;
  float* Sb = lds[ch][0];
  float* Tb = lds[ch][1];

  const int r  = lane & 15;
  const int hi = lane >> 4;

  // ---- embedding: product state u (wires0-3) x v (wires4-7) ----
  const float* xs = x + b * L_IN + ol;
  float cq[8], sq[8];
#pragma unroll
  for (int q = 0; q < 8; ++q) {
    float h = xs[q] * 1.57079632679489662f;   // (pi*x)/2
    cq[q] = __cosf(h);
    sq[q] = __sinf(h);
  }
  {
    // lane writes S0 entries e = lane*8 + j of col-major S (e = c*16 + r0):
    //   r0 = 8*(lane&1) + j (bits w0..w3), c = lane>>1 (bits w4..w7)
    int cidx = lane >> 1;
    float vc = ((cidx >> 3) & 1 ? sq[4] : cq[4]) *
               ((cidx >> 2) & 1 ? sq[5] : cq[5]) *
               ((cidx >> 1) & 1 ? sq[6] : cq[6]) *
               ((cidx      ) & 1 ? sq[7] : cq[7]);
    float w0f = (lane & 1) ? sq[0] : cq[0];
    float uv[8];
#pragma unroll
    for (int j = 0; j < 8; ++j) {
      uv[j] = w0f * ((j >> 2) & 1 ? sq[1] : cq[1]) *
                    ((j >> 1) & 1 ? sq[2] : cq[2]) *
                    ((j      ) & 1 ? sq[3] : cq[3]) * vc;
    }
    float4* s4 = (float4*)&Sb[lane * 8];
    s4[0] = make_float4(uv[0], uv[1], uv[2], uv[3]);
    s4[1] = make_float4(uv[4], uv[5], uv[6], uv[7]);
  }

  // ---- precompute CNOT-chain scatter addresses + parity signs ----
  // V (D-layout) lane/vgpr jv holds U[r][jv+8*hi] with U = post-gate state.
  // Full chain: r' = P(r); c' = P(c ^ 8*(r'&1))  (CNOT(3,4) between the halves)
  const int rp = cnot_perm4(r);
  int   addrp[8];
  float sgn[8];
#pragma unroll
  for (int jv = 0; jv < 8; ++jv) {
    int cc0 = (jv + 8 * hi) ^ ((rp & 1) << 3);
    int cp  = cnot_perm4(cc0);
    addrp[jv] = cp * 16 + rp;                          // col-major S index
    sgn[jv]   = ((__popc(rp) + __popc(cp)) & 1) ? -1.0f : 1.0f;
  }

  const v2f* packA = (const v2f*)pack;
  const int boff = r * 16 + 2 * hi;       // B-chunk base (same for S col-major / T row-major)

  float acc = 0.0f;
#pragma unroll
  for (int layer = 0; layer < 4; ++layer) {
    const int matA = ch * 8 + layer * 2;
    // T = K_A * S   (four chained 16x16x4 f32 WMMAs)
    v8f t = {};
#pragma unroll
    for (int q = 0; q < 4; ++q) {
      v2f a  = packA[(matA * 4 + q) * 32 + lane];
      v2f bv = *(const v2f*)&Sb[boff + 4 * q];
      t = __builtin_amdgcn_wmma_f32_16x16x4_f32(false, a, false, bv, (short)0, t,
                                                false, false);
    }
    // spill T (D-layout) to LDS row-major so T^T reads are contiguous b64
#pragma unroll
    for (int jv = 0; jv < 8; ++jv) Tb[(jv + 8 * hi) * 16 + r] = t[jv];

    // V = K_B * T^T  == (K_A S K_B^T)^T
    v8f vv = {};
#pragma unroll
    for (int q = 0; q < 4; ++q) {
      v2f a  = packA[((matA + 1) * 4 + q) * 32 + lane];
      v2f bv = *(const v2f*)&Tb[boff + 4 * q];
      vv = __builtin_amdgcn_wmma_f32_16x16x4_f32(false, a, false, bv, (short)0, vv,
                                                 false, false);
    }
    if (layer < 3) {
      // apply CNOT permutation while scattering back to S (col-major)
#pragma unroll
      for (int jv = 0; jv < 8; ++jv) Sb[addrp[jv]] = vv[jv];
    } else {
      // <Z..Z> = sum amp^2 * parity sign of the *permuted* index
#pragma unroll
      for (int jv = 0; jv < 8; ++jv) acc += vv[jv] * vv[jv] * sgn[jv];
    }
  }

  // wave32 reduction
#pragma unroll
  for (int off = 16; off > 0; off >>= 1) acc += __shfl_xor(acc, off, 32);
  if (lane == 0) out[(b * 8 + ch) * OUT_L + ol] = acc;
}

extern "C" void kernel_launch(void* const* d_in, const int* in_sizes, int n_in,
                              void* d_out, int out_size, void* d_ws, size_t ws_size,
                              hipStream_t stream) {
  const float* x = (const float*)d_in[0];        // (128,1,512) f32
  const float* w = (const float*)d_in[1];        // (8,32) f32
  float* out  = (float*)d_out;                   // (128,8,505) f32
  float* pack = (float*)d_ws;                    // 64 KB: 64 prepacked 16x16 f32 matrices

  qconv_pack_kernel<<<8, 256, 0, stream>>>(w, pack);
  qconv_main_kernel<<<dim3(OUT_L, 128), 256, 0, stream>>>(x, pack, out);
}